// LongformerAttentionMethod_60533269069845
// MI455X (gfx1250) — compile-verified
//
#include <hip/hip_runtime.h>
#include <hip/hip_bf16.h>

typedef float v2f __attribute__((ext_vector_type(2)));
typedef float v8f __attribute__((ext_vector_type(8)));

#define B_   2
#define H_   16
#define S_   4096
#define D_   64
#define G_   64
#define BH_  (B_ * H_)
#define KGPAD 66   // row stride (floats) for 64-wide LDS tiles: conflict-free, 8B aligned
#define PPAD  18   // row stride for 16-wide P staging tiles

static __device__ __forceinline__ v8f wmma4(v2f a, v2f b, v8f c) {
    // D = A(16x4 f32) * B(4x16 f32) + C(16x16 f32)
    return __builtin_amdgcn_wmma_f32_16x16x4_f32(false, a, false, b, (short)0, c,
                                                 false, false);
}

static __device__ __forceinline__ float negInf() { return -__builtin_inff(); }

// ---------------------------------------------------------------------------
// Kernel 0: compact global-token positions per batch via wave32 ballot scan.
// tp[b*G_+g] = token index of g-th global token, ng[b] = min(count, G_).
// ---------------------------------------------------------------------------
__global__ void __launch_bounds__(64)
build_token_pos(const int* __restrict__ mask, int* __restrict__ tp,
                int* __restrict__ ng) {
    const int b    = threadIdx.x >> 5;
    const int lane = threadIdx.x & 31;
    if (b >= B_) return;
    for (int g = lane; g < G_; g += 32) tp[b * G_ + g] = 0;
    int cnt = 0;
    for (int base = 0; base < S_; base += 32) {
        const int flag = mask[b * S_ + base + lane] != 0;
        const unsigned bal = (unsigned)__ballot(flag);
        const int prefix = __popc(bal & ((1u << lane) - 1u));
        if (flag && (cnt + prefix) < G_) tp[b * G_ + cnt + prefix] = base + lane;
        cnt += __popc(bal);
    }
    if (lane == 0) ng[b] = cnt < G_ ? cnt : G_;
}

// ---------------------------------------------------------------------------
// Kernel A: every query attends over the compacted global KV (G_=64 slots).
// grid (BH_, S_/64), block 128 (4 waves, 16 query rows each).
// Writes probs[bh,s,G_] and context[bh,s,D_] for ALL rows.
// ---------------------------------------------------------------------------
__global__ void __launch_bounds__(128)
partA_global_kv_attn(const float* __restrict__ q, const float* __restrict__ k,
                     const float* __restrict__ v, const int* __restrict__ tp,
                     const int* __restrict__ ngp, float* __restrict__ ctx,
                     float* __restrict__ probs) {
    __shared__ float Kg[G_ * KGPAD];
    __shared__ float Vg[G_ * KGPAD];
    __shared__ float Pst[4 * 16 * KGPAD];

    const int bh    = blockIdx.x;
    const int b     = bh / H_;
    const int tid   = threadIdx.x;
    const int wave  = tid >> 5;
    const int lane  = tid & 31;
    const int ln    = lane & 15;
    const int h     = lane >> 4;
    const int qbase = blockIdx.y * 64 + wave * 16;
    const int ng    = ngp[b];

    // stage compacted global K/V into LDS (zeros for invalid slots)
    for (int idx = tid; idx < G_ * 32; idx += 128) {
        const int row = idx >> 5;
        const int cp  = (idx & 31) << 1;
        v2f kk2 = {0.f, 0.f}, vv2 = {0.f, 0.f};
        if (row < ng) {
            const size_t base = ((size_t)bh * S_ + tp[b * G_ + row]) * D_ + cp;
            kk2 = *(const v2f*)(k + base);
            vv2 = *(const v2f*)(v + base);
        }
        *(v2f*)(Kg + row * KGPAD + cp) = kk2;
        *(v2f*)(Vg + row * KGPAD + cp) = vv2;
    }
    __syncthreads();

    // preload Q A-fragments: lane ln = row m, VGPR pair covers K = 4kk+2h, +1
    const float* qrow = q + ((size_t)bh * S_ + (qbase + ln)) * D_;
    v2f qa[16];
#pragma unroll
    for (int kk = 0; kk < 16; ++kk)
        qa[kk] = *(const v2f*)(qrow + 4 * kk + 2 * h);

    // scores (16 x 64) = Q x Kg^T, 4 N-tiles of 16
    v8f cs[4];
#pragma unroll
    for (int nt = 0; nt < 4; ++nt) {
        v8f c = {0.f, 0.f, 0.f, 0.f, 0.f, 0.f, 0.f, 0.f};
        const float* kgp = Kg + (nt * 16 + ln) * KGPAD;  // B[d][g]: N = g = nt*16+ln
#pragma unroll
        for (int kk = 0; kk < 16; ++kk) {
            const v2f bfr = *(const v2f*)(kgp + 4 * kk + 2 * h);
            c = wmma4(qa[kk], bfr, c);
        }
        cs[nt] = c;
    }

    // scale by 1/sqrt(D) and mask invalid global slots
    const float inv_scale = 0.125f;
#pragma unroll
    for (int nt = 0; nt < 4; ++nt) {
        const bool invalid = (nt * 16 + ln) >= ng;
#pragma unroll
        for (int r = 0; r < 8; ++r) {
            const float s = cs[nt][r] * inv_scale;
            cs[nt][r] = invalid ? negInf() : s;
        }
    }

    // fp32 softmax over the 64 global slots (row lives across 16 lanes)
    float linv[8];
#pragma unroll
    for (int r = 0; r < 8; ++r) {
        float m = fmaxf(fmaxf(cs[0][r], cs[1][r]), fmaxf(cs[2][r], cs[3][r]));
#pragma unroll
        for (int off = 1; off < 16; off <<= 1) m = fmaxf(m, __shfl_xor(m, off, 16));
        float ssum = 0.f;
#pragma unroll
        for (int nt = 0; nt < 4; ++nt) {
            const float p = __expf(cs[nt][r] - m);
            cs[nt][r] = p;
            ssum += p;
        }
#pragma unroll
        for (int off = 1; off < 16; off <<= 1) ssum += __shfl_xor(ssum, off, 16);
        linv[r] = 1.f / ssum;
    }

    // write probs and stage P (C-layout -> LDS for A-layout reload)
    float* pst = Pst + wave * 16 * KGPAD;
#pragma unroll
    for (int nt = 0; nt < 4; ++nt) {
#pragma unroll
        for (int r = 0; r < 8; ++r) {
            const float p  = cs[nt][r] * linv[r];
            const int  row = r + 8 * h;
            probs[((size_t)bh * S_ + qbase + row) * G_ + nt * 16 + ln] = p;
            pst[row * KGPAD + nt * 16 + ln] = p;
        }
    }

    // context (16 x 64) = P (16 x 64) x Vg (64 x 64)
#pragma unroll
    for (int nt = 0; nt < 4; ++nt) {
        v8f acc = {0.f, 0.f, 0.f, 0.f, 0.f, 0.f, 0.f, 0.f};
#pragma unroll
        for (int kk = 0; kk < 16; ++kk) {
            const v2f a = *(const v2f*)(pst + ln * KGPAD + 4 * kk + 2 * h);
            v2f bfr;
            bfr.x = Vg[(4 * kk + 2 * h) * KGPAD + nt * 16 + ln];
            bfr.y = Vg[(4 * kk + 2 * h + 1) * KGPAD + nt * 16 + ln];
            acc = wmma4(a, bfr, acc);
        }
#pragma unroll
        for (int r = 0; r < 8; ++r)
            ctx[((size_t)bh * S_ + qbase + r + 8 * h) * D_ + nt * 16 + ln] = acc[r];
    }
}

// ---------------------------------------------------------------------------
// Kernel B: global queries do full attention over all S keys (flash-style).
// grid (BH_, G_/16), block 256 (8 waves split S into 512-key segments, then
// LDS combine). Overwrites context rows at the global token positions.
// ---------------------------------------------------------------------------
__global__ void __launch_bounds__(256)
partB_full_attn(const float* __restrict__ q, const float* __restrict__ k,
                const float* __restrict__ v, const int* __restrict__ tp,
                const int* __restrict__ ngp, float* __restrict__ ctx) {
    __shared__ float PstB[8 * 16 * PPAD];
    __shared__ float AccSm[8 * 16 * D_];
    __shared__ float Msm[8 * 16];
    __shared__ float Lsm[8 * 16];
    __shared__ int   stok[16];

    const int bh   = blockIdx.x;
    const int b    = bh / H_;
    const int qt   = blockIdx.y;
    const int tid  = threadIdx.x;
    const int wave = tid >> 5;
    const int lane = tid & 31;
    const int ln   = lane & 15;
    const int h    = lane >> 4;
    const int ng   = ngp[b];

    if (tid < 16) {
        const int g = qt * 16 + tid;
        stok[tid] = (g < ng) ? tp[b * G_ + g] : 0;
    }
    __syncthreads();

    // Q fragments for the 16 global queries of this tile
    const float* qrow = q + ((size_t)bh * S_ + stok[ln]) * D_;
    v2f qa[16];
#pragma unroll
    for (int kk = 0; kk < 16; ++kk)
        qa[kk] = *(const v2f*)(qrow + 4 * kk + 2 * h);

    float mrow[8], lrow[8], corr[8];
    v8f acc[4];
#pragma unroll
    for (int r = 0; r < 8; ++r) { mrow[r] = negInf(); lrow[r] = 0.f; }
#pragma unroll
    for (int nt = 0; nt < 4; ++nt) acc[nt] = (v8f){0.f,0.f,0.f,0.f,0.f,0.f,0.f,0.f};

    float* pst = PstB + wave * 16 * PPAD;
    const int kstart = wave * (S_ / 8);

    for (int ch = 0; ch < (S_ / 8) / 16; ++ch) {  // 32 chunks of 16 keys
        const int kb = kstart + ch * 16;
        const float* kp = k + ((size_t)bh * S_ + kb) * D_;

        // scores tile (16 queries x 16 keys)
        v8f c = {0.f, 0.f, 0.f, 0.f, 0.f, 0.f, 0.f, 0.f};
#pragma unroll
        for (int kk = 0; kk < 16; ++kk) {
            const v2f bfr = *(const v2f*)(kp + ln * D_ + 4 * kk + 2 * h);
            c = wmma4(qa[kk], bfr, c);
        }

        // online softmax update
#pragma unroll
        for (int r = 0; r < 8; ++r) {
            const float s = c[r] * 0.125f;
            float m = s;
#pragma unroll
            for (int off = 1; off < 16; off <<= 1) m = fmaxf(m, __shfl_xor(m, off, 16));
            const float m2 = fmaxf(mrow[r], m);
            corr[r] = __expf(mrow[r] - m2);
            const float p = __expf(s - m2);
            c[r] = p;
            float ssum = p;
#pragma unroll
            for (int off = 1; off < 16; off <<= 1) ssum += __shfl_xor(ssum, off, 16);
            lrow[r] = lrow[r] * corr[r] + ssum;
            mrow[r] = m2;
            pst[(r + 8 * h) * PPAD + ln] = p;  // stage P for A-layout reload
        }

        // acc = acc * corr + P (16x16) x V_chunk (16x64)
#pragma unroll
        for (int nt = 0; nt < 4; ++nt) {
#pragma unroll
            for (int r = 0; r < 8; ++r) acc[nt][r] *= corr[r];
#pragma unroll
            for (int kk = 0; kk < 4; ++kk) {
                const v2f a = *(const v2f*)(pst + ln * PPAD + 4 * kk + 2 * h);
                v2f bfr;
                const float* vp = v + ((size_t)bh * S_ + kb + 4 * kk + 2 * h) * D_
                                  + nt * 16 + ln;
                bfr.x = vp[0];
                bfr.y = vp[D_];
                acc[nt] = wmma4(a, bfr, acc[nt]);
            }
        }
    }

    // dump per-wave partials and combine across the 8 S-segments
#pragma unroll
    for (int nt = 0; nt < 4; ++nt)
#pragma unroll
        for (int r = 0; r < 8; ++r)
            AccSm[(wave * 16 + r + 8 * h) * D_ + nt * 16 + ln] = acc[nt][r];
    if (ln == 0) {
#pragma unroll
        for (int r = 0; r < 8; ++r) {
            Msm[wave * 16 + r + 8 * h] = mrow[r];
            Lsm[wave * 16 + r + 8 * h] = lrow[r];
        }
    }
    __syncthreads();

    for (int o = tid; o < 16 * D_; o += 256) {
        const int row = o >> 6;
        const int col = o & 63;
        float M = negInf();
        for (int w = 0; w < 8; ++w) M = fmaxf(M, Msm[w * 16 + row]);
        float L = 0.f, cx = 0.f;
        for (int w = 0; w < 8; ++w) {
            const float e = __expf(Msm[w * 16 + row] - M);
            L  += Lsm[w * 16 + row] * e;
            cx += AccSm[(w * 16 + row) * D_ + col] * e;
        }
        const int g = qt * 16 + row;
        if (g < ng)
            ctx[((size_t)bh * S_ + stok[row]) * D_ + col] = cx / L;
    }
}

// ---------------------------------------------------------------------------
extern "C" void kernel_launch(void* const* d_in, const int* in_sizes, int n_in,
                              void* d_out, int out_size, void* d_ws, size_t ws_size,
                              hipStream_t stream) {
    (void)in_sizes; (void)n_in; (void)out_size; (void)ws_size;
    const float* q    = (const float*)d_in[0];
    const float* k    = (const float*)d_in[1];
    const float* v    = (const float*)d_in[2];
    const int*   mask = (const int*)d_in[3];
    // d_in[4] = max_global (== G_ compile-time)

    int* tp = (int*)d_ws;          // B_*G_ token positions
    int* ng = tp + B_ * G_;        // B_ valid-count

    float* ctx   = (float*)d_out;
    float* probs = ctx + (size_t)B_ * H_ * S_ * D_;

    build_token_pos<<<1, 64, 0, stream>>>(mask, tp, ng);

    dim3 gridA(BH_, S_ / 64);
    partA_global_kv_attn<<<gridA, 128, 0, stream>>>(q, k, v, tp, ng, ctx, probs);

    dim3 gridB(BH_, G_ / 16);
    partB_full_attn<<<gridB, 256, 0, stream>>>(q, k, v, tp, ng, ctx);
}